// IFNeuron_90048284328050
// MI455X (gfx1250) — compile-verified
//
#include <hip/hip_runtime.h>
#include <cstdint>
#include <cstddef>

// ---------------------------------------------------------------------------
// IF neuron forward: out[t] = spike(v + x[t]), v <- v + x[t] - spike(...)
// x: [T=4, B, H, W] f32, out: same shape f32.
// Pure streaming problem: ~0.54 GB moved -> ~23 us floor at 23.3 TB/s HBM.
// Strategy: b128 loads/stores, CDNA5 async global->LDS double-buffered
// pipeline (ASYNCcnt), non-temporal streaming stores. No WMMA (no matmul).
// ---------------------------------------------------------------------------

#define TH_V    1.0f
#define T_STEPS 4
#define BLOCK   256
#define MAX_BLOCKS 4096

#ifndef __has_builtin
#define __has_builtin(x) 0
#endif

#if defined(__HIP_DEVICE_COMPILE__) && defined(__gfx1250__) && \
    __has_builtin(__builtin_amdgcn_global_load_async_to_lds_b128) && \
    __has_builtin(__builtin_amdgcn_s_wait_asynccnt)
#define USE_ASYNC 1
#else
#define USE_ASYNC 0
#endif

typedef float v4f __attribute__((ext_vector_type(4)));
typedef int   v4i_n __attribute__((vector_size(16)));   // builtin's operand type

__device__ __forceinline__ v4f if_step(v4f xt, v4f& v) {
  v4f o;
  v.x += xt.x; o.x = (v.x >= TH_V) ? TH_V : 0.0f; v.x -= o.x;
  v.y += xt.y; o.y = (v.y >= TH_V) ? TH_V : 0.0f; v.y -= o.y;
  v.z += xt.z; o.z = (v.z >= TH_V) ? TH_V : 0.0f; v.z -= o.z;
  v.w += xt.w; o.w = (v.w >= TH_V) ? TH_V : 0.0f; v.w -= o.w;
  return o;
}

// Main vectorized kernel. nvec = elements-per-timestep / 4 (float4 units).
// Each "tile" = BLOCK float4 per timestep. Grid-stride over tiles.
__global__ void __launch_bounds__(BLOCK)
if_neuron_main(const float* __restrict__ x, float* __restrict__ out,
               size_t nvec, size_t ntiles) {
  const int tid = threadIdx.x;
  const size_t stride = gridDim.x;
  const v4f* __restrict__ xv = reinterpret_cast<const v4f*>(x);
  v4f* __restrict__ ov = reinterpret_cast<v4f*>(out);

#if USE_ASYNC
  // Double-buffered async global->LDS staging.
  // 2 bufs x 4 timesteps x 256 lanes x 16B = 32 KB (<< 320 KB per WGP).
  __shared__ v4f smem[2][T_STEPS][BLOCK];

  auto issue = [&](size_t tl, int b) {
    const size_t base = tl * (size_t)BLOCK + (size_t)tid;
#pragma unroll
    for (int t = 0; t < T_STEPS; ++t) {
      const v4f* g = xv + (size_t)t * nvec + base;
      // Builtin signature: (v4i addrspace(1)*, v4i addrspace(3)*, Ii, Ii).
      // Global pointer -> AS1 via inttoptr; LDS slot -> AS3 via the
      // flat-address low 32 bits (== LDS offset per the aperture mapping).
      auto gp = (__attribute__((address_space(1))) v4i_n*)(uintptr_t)g;
      auto lp = (__attribute__((address_space(3))) v4i_n*)
                    (uint32_t)(uintptr_t)&smem[b][t][tid];
      __builtin_amdgcn_global_load_async_to_lds_b128(gp, lp, /*offset=*/0,
                                                     /*cpol=*/0);
    }
  };

  size_t tl = blockIdx.x;
  if (tl >= ntiles) return;
  issue(tl, 0);                       // prime the pipeline (ASYNCcnt = 4)
  int buf = 0;
  for (; tl < ntiles; tl += stride) {
    const size_t nx = tl + stride;
    if (nx < ntiles) {
      issue(nx, buf ^ 1);             // ASYNCcnt -> 8
      // Async loads complete in order: cnt<=4 releases the current buffer
      // while the next tile's 4 loads are still in flight.
      __builtin_amdgcn_s_wait_asynccnt(T_STEPS);
    } else {
      __builtin_amdgcn_s_wait_asynccnt(0);
    }
    v4f v = {0.0f, 0.0f, 0.0f, 0.0f};
    const size_t obase = tl * (size_t)BLOCK + (size_t)tid;
#pragma unroll
    for (int t = 0; t < T_STEPS; ++t) {
      v4f o = if_step(smem[buf][t][tid], v);    // ds_load_b128
      __builtin_nontemporal_store(o, ov + (size_t)t * nvec + obase);
    }
    buf ^= 1;
  }
#else
  // Fallback: direct b128 streaming loads (still bandwidth-optimal).
  for (size_t tl = blockIdx.x; tl < ntiles; tl += stride) {
    const size_t base = tl * (size_t)BLOCK + (size_t)tid;
    v4f v = {0.0f, 0.0f, 0.0f, 0.0f};
#pragma unroll
    for (int t = 0; t < T_STEPS; ++t) {
      v4f xt = __builtin_nontemporal_load(xv + (size_t)t * nvec + base);
      v4f o = if_step(xt, v);
      __builtin_nontemporal_store(o, ov + (size_t)t * nvec + base);
    }
  }
#endif
}

// Scalar tail kernel: covers elements [start, n) of each timestep.
__global__ void if_neuron_tail(const float* __restrict__ x,
                               float* __restrict__ out,
                               size_t n, size_t start) {
  size_t i = start + (size_t)blockIdx.x * blockDim.x + threadIdx.x;
  if (i >= n) return;
  float v = 0.0f;
#pragma unroll
  for (int t = 0; t < T_STEPS; ++t) {
    v += x[(size_t)t * n + i];
    float o = (v >= TH_V) ? TH_V : 0.0f;
    v -= o;
    out[(size_t)t * n + i] = o;
  }
}

extern "C" void kernel_launch(void* const* d_in, const int* in_sizes, int n_in,
                              void* d_out, int out_size, void* d_ws,
                              size_t ws_size, hipStream_t stream) {
  (void)n_in; (void)out_size; (void)d_ws; (void)ws_size;
  const float* x = (const float*)d_in[0];
  float* out = (float*)d_out;

  const size_t total = (size_t)in_sizes[0];
  const size_t n = total / (size_t)T_STEPS;   // elements per timestep

  if ((n % 4) == 0) {
    const size_t nvec = n / 4;
    const size_t ntiles = nvec / (size_t)BLOCK;
    if (ntiles) {
      const size_t blocks = ntiles < (size_t)MAX_BLOCKS ? ntiles
                                                        : (size_t)MAX_BLOCKS;
      if_neuron_main<<<(dim3)(unsigned)blocks, BLOCK, 0, stream>>>(x, out,
                                                                  nvec, ntiles);
    }
    const size_t done = ntiles * (size_t)BLOCK * 4;
    if (done < n) {
      const size_t rem = n - done;
      const unsigned tg = (unsigned)((rem + 255) / 256);
      if_neuron_tail<<<tg, 256, 0, stream>>>(x, out, n, done);
    }
  } else {
    const unsigned tg = (unsigned)((n + 255) / 256);
    if_neuron_tail<<<tg, 256, 0, stream>>>(x, out, n, 0);
  }
}